// Model_84387517432373
// MI455X (gfx1250) — compile-verified
//
#include <hip/hip_runtime.h>

typedef float v2f __attribute__((ext_vector_type(2)));
typedef float v8f __attribute__((ext_vector_type(8)));
typedef int   v4i_vs __attribute__((vector_size(16)));
typedef __attribute__((address_space(1))) v4i_vs* gptr_v4i;
typedef __attribute__((address_space(3))) v4i_vs* lptr_v4i;

#define NROW   2048        // B * N_AGENTS
#define NBATCH 32
#define NA     64
#define TXX    20
#define TYY    40
#define HID    48
#define G3     144         // 3*HID
#define CE     16
#define FVC    16
#define SPK    36
#define SPD    1728        // SPK*HID
#define KSAMP  4
#define HZ_F   10.0f
#define TWO_PI_F 6.28318530717958647692f
#define R_STEP_F (3.5f/6.0f)
#define T_STEP_F (TWO_PI_F/6.0f)

#if defined(__has_builtin)
#if __has_builtin(__builtin_amdgcn_global_load_async_to_lds_b128)
#define HAVE_ASYNC_LDS 1
#endif
#endif
#ifndef HAVE_ASYNC_LDS
#define HAVE_ASYNC_LDS 0
#endif

#define WMMA_F32(a, b, c) \
    __builtin_amdgcn_wmma_f32_16x16x4_f32(false, (a), false, (b), (short)0, (c), false, false)

#if HAVE_ASYNC_LDS
__device__ __forceinline__ void async_cp16(const float* g, float* l)
{
    __builtin_amdgcn_global_load_async_to_lds_b128(
        (gptr_v4i)(unsigned long long)g,
        (lptr_v4i)(unsigned int)(unsigned long long)(__builtin_amdgcn_is_shared(l) ? l : l),
        0, 0);
}
#endif

// ---------------------------------------------------------------------------
// Generic fp32 WMMA GEMM: C[M x N] = A[M x K] * W[N x K]^T + bias, + activation
// One wave per 16x16 output tile; K must be a multiple of 8.
// Depth-2 register pipeline so loads of chunk k+1/k+2 overlap the WMMA on k.
// act: 0 = none, 1 = relu, 2 = 0.5*exp
// ---------------------------------------------------------------------------
__global__ __launch_bounds__(32)
void gemm_wmma_f32(const float* __restrict__ A, int lda,
                   const float* __restrict__ W, int ldw,
                   const float* __restrict__ bias,
                   float* __restrict__ C, int ldc, int K, int act)
{
    int lane = threadIdx.x;
    int half = lane >> 4;
    int l16  = lane & 15;
    int row0 = blockIdx.x * 16;
    int col0 = blockIdx.y * 16;

    const float* ap = A + (size_t)(row0 + l16) * lda + 2 * half;
    const float* wp = W + (size_t)(col0 + l16) * ldw + 2 * half;

    v8f acc = {};
    v2f a0 = *(const v2f*)(ap + 0);
    v2f b0 = *(const v2f*)(wp + 0);
    v2f a1 = *(const v2f*)(ap + 4);
    v2f b1 = *(const v2f*)(wp + 4);
    int k = 8;
    for (; k + 8 <= K; k += 8) {
        v2f a2 = *(const v2f*)(ap + k);
        v2f b2 = *(const v2f*)(wp + k);
        v2f a3 = *(const v2f*)(ap + k + 4);
        v2f b3 = *(const v2f*)(wp + k + 4);
        acc = WMMA_F32(a0, b0, acc);
        acc = WMMA_F32(a1, b1, acc);
        a0 = a2; b0 = b2; a1 = a3; b1 = b3;
    }
    acc = WMMA_F32(a0, b0, acc);
    acc = WMMA_F32(a1, b1, acc);

    float bv = bias ? bias[col0 + l16] : 0.0f;
#pragma unroll
    for (int i = 0; i < 8; ++i) {
        int r = row0 + i + 8 * half;
        float v = acc[i] + bv;
        if (act == 1)      v = fmaxf(v, 0.0f);
        else if (act == 2) v = 0.5f * expf(v);
        C[(size_t)r * ldc + col0 + l16] = v;
    }
}

// ---------------------------------------------------------------------------
// Specialized WMMA GEMM for the social-pooling FC (K = 1728 = 27*64):
// stages 16x64 chunks of A and W in LDS via async global->LDS (double
// buffered, tracked with ASYNCcnt), then feeds WMMA from LDS.
// LDS pitch 68 floats: row stride = 4 banks mod 64 -> conflict-free b64 reads.
// Epilogue: bias + relu, strided output (ldc).
// ---------------------------------------------------------------------------
#define FCS_KC 64
#define LPITCH 68

__global__ __launch_bounds__(32)
void gemm_fcs_wmma(const float* __restrict__ A, int lda,
                   const float* __restrict__ W, int ldw,
                   const float* __restrict__ bias,
                   float* __restrict__ C, int ldc, int K)
{
    __shared__ float sA[2][16 * LPITCH];
    __shared__ float sW[2][16 * LPITCH];

    int lane = threadIdx.x;
    int half = lane >> 4;
    int l16  = lane & 15;
    int row0 = blockIdx.x * 16;
    int col0 = blockIdx.y * 16;

    // per-chunk copy: 8 instructions x 32 lanes x 16B cover 16 rows x 256B
    int crow = lane >> 4;        // +2 per i
    int cseg = lane & 15;        // 16B segment within row

    int nch = K / FCS_KC;
    v8f acc = {};

#if HAVE_ASYNC_LDS
    // ---- async double-buffered pipeline ----
    {
        // prologue: chunk 0 -> buf 0
#pragma unroll
        for (int i = 0; i < 8; ++i) {
            int r = i * 2 + crow;
            async_cp16(A + (size_t)(row0 + r) * lda + cseg * 4,
                       &sA[0][r * LPITCH + cseg * 4]);
            async_cp16(W + (size_t)(col0 + r) * ldw + cseg * 4,
                       &sW[0][r * LPITCH + cseg * 4]);
        }
    }
    for (int c = 0; c < nch; ++c) {
        int buf = c & 1;
        if (c + 1 < nch) {
            int kc = (c + 1) * FCS_KC;
#pragma unroll
            for (int i = 0; i < 8; ++i) {
                int r = i * 2 + crow;
                async_cp16(A + (size_t)(row0 + r) * lda + kc + cseg * 4,
                           &sA[buf ^ 1][r * LPITCH + cseg * 4]);
                async_cp16(W + (size_t)(col0 + r) * ldw + kc + cseg * 4,
                           &sW[buf ^ 1][r * LPITCH + cseg * 4]);
            }
            // chunk c complete once only the 16 newer transfers remain
            asm volatile("s_wait_asynccnt 0x10" ::: "memory");
        } else {
            asm volatile("s_wait_asynccnt 0x0" ::: "memory");
        }
        const float* la = &sA[buf][l16 * LPITCH + 2 * half];
        const float* lw = &sW[buf][l16 * LPITCH + 2 * half];
#pragma unroll
        for (int kk = 0; kk < FCS_KC; kk += 4) {
            v2f a = *(const v2f*)(la + kk);
            v2f b = *(const v2f*)(lw + kk);
            acc = WMMA_F32(a, b, acc);
        }
    }
#else
    // ---- synchronous LDS-staged fallback ----
    for (int c = 0; c < nch; ++c) {
        int kc = c * FCS_KC;
        __syncthreads();
#pragma unroll
        for (int i = 0; i < 8; ++i) {
            int r = i * 2 + crow;
            const float4 va = *(const float4*)(A + (size_t)(row0 + r) * lda + kc + cseg * 4);
            const float4 vw = *(const float4*)(W + (size_t)(col0 + r) * ldw + kc + cseg * 4);
            *(float4*)&sA[0][r * LPITCH + cseg * 4] = va;
            *(float4*)&sW[0][r * LPITCH + cseg * 4] = vw;
        }
        __syncthreads();
        const float* la = &sA[0][l16 * LPITCH + 2 * half];
        const float* lw = &sW[0][l16 * LPITCH + 2 * half];
#pragma unroll
        for (int kk = 0; kk < FCS_KC; kk += 4) {
            v2f a = *(const v2f*)(la + kk);
            v2f b = *(const v2f*)(lw + kk);
            acc = WMMA_F32(a, b, acc);
        }
    }
#endif

    float bv = bias[col0 + l16];
#pragma unroll
    for (int i = 0; i < 8; ++i) {
        int r = row0 + i + 8 * half;
        C[(size_t)r * ldc + col0 + l16] = fmaxf(acc[i] + bv, 0.0f);
    }
}

// ---------------------------------------------------------------------------
// Fused GRU step: gates via WMMA (9 waves: one per 16-wide column tile of the
// 144-wide gate matrix), then sigmoid/tanh update. Register-pipelined K loops.
//   gx = x @ wi^T (K=xk, x==nullptr -> 0); gh = h @ wh^T (K=HID, hin==nullptr -> 0)
// ---------------------------------------------------------------------------
__global__ __launch_bounds__(288)
void gru_step_wmma(const float* __restrict__ x, int xk,
                   const float* __restrict__ hin,
                   const float* __restrict__ wi, const float* __restrict__ wh,
                   const float* __restrict__ bi, const float* __restrict__ bh,
                   float* __restrict__ hout)
{
    __shared__ float s_gx[16 * G3];
    __shared__ float s_gh[16 * G3];

    int tid  = threadIdx.x;
    int wave = tid >> 5;
    int lane = tid & 31;
    int half = lane >> 4;
    int l16  = lane & 15;
    int r0   = blockIdx.x * 16;
    int c0   = wave * 16;

    v8f ax = {};
    if (x) {
        const float* ap = x  + (size_t)(r0 + l16) * xk + 2 * half;
        const float* wp = wi + (size_t)(c0 + l16) * xk + 2 * half;
        v2f a0 = *(const v2f*)(ap + 0), b0 = *(const v2f*)(wp + 0);
        v2f a1 = *(const v2f*)(ap + 4), b1 = *(const v2f*)(wp + 4);
        int k = 8;
        for (; k + 8 <= xk; k += 8) {
            v2f a2 = *(const v2f*)(ap + k),     b2 = *(const v2f*)(wp + k);
            v2f a3 = *(const v2f*)(ap + k + 4), b3 = *(const v2f*)(wp + k + 4);
            ax = WMMA_F32(a0, b0, ax);
            ax = WMMA_F32(a1, b1, ax);
            a0 = a2; b0 = b2; a1 = a3; b1 = b3;
        }
        ax = WMMA_F32(a0, b0, ax);
        ax = WMMA_F32(a1, b1, ax);
    }
    v8f ah = {};
    if (hin) {
        const float* ap = hin + (size_t)(r0 + l16) * HID + 2 * half;
        const float* wp = wh  + (size_t)(c0 + l16) * HID + 2 * half;
        v2f a0 = *(const v2f*)(ap + 0), b0 = *(const v2f*)(wp + 0);
        v2f a1 = *(const v2f*)(ap + 4), b1 = *(const v2f*)(wp + 4);
        int k = 8;
        for (; k + 8 <= HID; k += 8) {
            v2f a2 = *(const v2f*)(ap + k),     b2 = *(const v2f*)(wp + k);
            v2f a3 = *(const v2f*)(ap + k + 4), b3 = *(const v2f*)(wp + k + 4);
            ah = WMMA_F32(a0, b0, ah);
            ah = WMMA_F32(a1, b1, ah);
            a0 = a2; b0 = b2; a1 = a3; b1 = b3;
        }
        ah = WMMA_F32(a0, b0, ah);
        ah = WMMA_F32(a1, b1, ah);
    }
#pragma unroll
    for (int i = 0; i < 8; ++i) {
        int r = i + 8 * half;
        s_gx[r * G3 + c0 + l16] = ax[i];
        s_gh[r * G3 + c0 + l16] = ah[i];
    }
    __syncthreads();

    for (int idx = tid; idx < 16 * HID; idx += 288) {
        int row = idx / HID;
        int c   = idx % HID;
        float xr = s_gx[row * G3 + c]           + bi[c];
        float xz = s_gx[row * G3 + HID + c]     + bi[HID + c];
        float xn = s_gx[row * G3 + 2 * HID + c] + bi[2 * HID + c];
        float hr = s_gh[row * G3 + c]           + bh[c];
        float hz = s_gh[row * G3 + HID + c]     + bh[HID + c];
        float hn = s_gh[row * G3 + 2 * HID + c] + bh[2 * HID + c];
        float rg = 1.0f / (1.0f + expf(-(xr + hr)));
        float zg = 1.0f / (1.0f + expf(-(xz + hz)));
        float ng = tanhf(xn + rg * hn);
        float hp = hin ? hin[(size_t)(r0 + row) * HID + c] : 0.0f;
        hout[(size_t)(r0 + row) * HID + c] = (1.0f - zg) * ng + zg * hp;
    }
}

// ---------------------------------------------------------------------------
// conv1: img (32,4,160,160) -> relu -> fm1 (32,16,160,160), 3x3 pad1 stride1
// ---------------------------------------------------------------------------
__global__ void conv1_kernel(const float* __restrict__ img,
                             const float* __restrict__ w, const float* __restrict__ b,
                             float* __restrict__ fm1)
{
    int idx = blockIdx.x * blockDim.x + threadIdx.x;
    if (idx >= NBATCH * 160 * 160) return;
    int ox = idx % 160, oy = (idx / 160) % 160, bb = idx / (160 * 160);
    float in[4][9];
#pragma unroll
    for (int ic = 0; ic < 4; ++ic)
#pragma unroll
        for (int t = 0; t < 9; ++t) {
            int iy = oy + t / 3 - 1, ix = ox + t % 3 - 1;
            in[ic][t] = (iy >= 0 && iy < 160 && ix >= 0 && ix < 160)
                      ? img[(((size_t)bb * 4 + ic) * 160 + iy) * 160 + ix] : 0.0f;
        }
    for (int oc = 0; oc < 16; ++oc) {
        float acc = b[oc];
#pragma unroll
        for (int ic = 0; ic < 4; ++ic)
#pragma unroll
            for (int t = 0; t < 9; ++t)
                acc += in[ic][t] * w[(oc * 4 + ic) * 9 + t];
        fm1[(((size_t)bb * 16 + oc) * 160 + oy) * 160 + ox] = fmaxf(acc, 0.0f);
    }
}

// ---------------------------------------------------------------------------
// conv2: fm1 (32,16,160,160) -> stride2 relu -> fm_t (32,80,80,32)  (NHWC)
// ---------------------------------------------------------------------------
__global__ void conv2_kernel(const float* __restrict__ fm1,
                             const float* __restrict__ w, const float* __restrict__ b,
                             float* __restrict__ fmt)
{
    int idx = blockIdx.x * blockDim.x + threadIdx.x;
    if (idx >= NBATCH * 80 * 80) return;
    int ox = idx % 80, oy = (idx / 80) % 80, bb = idx / (80 * 80);
    float acc[32];
#pragma unroll
    for (int oc = 0; oc < 32; ++oc) acc[oc] = b[oc];
    for (int ky = 0; ky < 3; ++ky) {
        int iy = oy * 2 + ky - 1;
        if (iy < 0 || iy >= 160) continue;
        for (int kx = 0; kx < 3; ++kx) {
            int ix = ox * 2 + kx - 1;
            if (ix < 0 || ix >= 160) continue;
            float inv[16];
#pragma unroll
            for (int ic = 0; ic < 16; ++ic)
                inv[ic] = fm1[(((size_t)bb * 16 + ic) * 160 + iy) * 160 + ix];
#pragma unroll
            for (int oc = 0; oc < 32; ++oc) {
                float s = 0.0f;
#pragma unroll
                for (int ic = 0; ic < 16; ++ic)
                    s += inv[ic] * w[((oc * 16 + ic) * 3 + ky) * 3 + kx];
                acc[oc] += s;
            }
        }
    }
    for (int oc = 0; oc < 32; ++oc)
        fmt[(((size_t)bb * 80 + oy) * 80 + ox) * 32 + oc] = fmaxf(acc[oc], 0.0f);
}

// conv1d encoder: tr (NROW,2,T) -> relu -> ex (T, NROW, CE)
__global__ void enc1d_kernel(const float* __restrict__ tr, int T,
                             const float* __restrict__ w, const float* __restrict__ b,
                             float* __restrict__ ex)
{
    int idx = blockIdx.x * blockDim.x + threadIdx.x;
    if (idx >= T * NROW * CE) return;
    int c = idx % CE, n = (idx / CE) % NROW, t = idx / (CE * NROW);
    float acc = b[c];
#pragma unroll
    for (int ic = 0; ic < 2; ++ic)
#pragma unroll
        for (int dt = 0; dt < 3; ++dt) {
            int tt = t + dt - 1;
            if (tt >= 0 && tt < T)
                acc += tr[((size_t)n * 2 + ic) * T + tt] * w[(c * 2 + ic) * 3 + dt];
        }
    ex[((size_t)t * NROW + n) * CE + c] = fmaxf(acc, 0.0f);
}

__global__ void cur_kernel(const float* __restrict__ tx, float* __restrict__ cur)
{
    int idx = blockIdx.x * blockDim.x + threadIdx.x;
    if (idx >= NROW * 2) return;
    int n = idx / 2, c = idx % 2;
    cur[n * 2 + c] = tx[((size_t)n * 2 + c) * TXX + (TXX - 1)];
}

__global__ void concat_kernel(const float* __restrict__ a, const float* __restrict__ b,
                              float* __restrict__ o)
{
    int idx = blockIdx.x * blockDim.x + threadIdx.x;
    if (idx >= NROW * 96) return;
    int n = idx / 96, c = idx % 96;
    o[idx] = (c < HID) ? a[(size_t)n * HID + c] : b[(size_t)n * HID + c - HID];
}

__device__ inline unsigned pcg_hash(unsigned v)
{
    v = v * 747796405u + 2891336453u;
    unsigned w = ((v >> ((v >> 28) + 4u)) ^ v) * 277803737u;
    return (w >> 22) ^ w;
}

// z = H_delta * normal(hash) + H_miu   (deterministic stand-in for jax PRNG)
__global__ void znoise_kernel(const float* __restrict__ miu, const float* __restrict__ dlt,
                              float* __restrict__ z, int samp)
{
    int idx = blockIdx.x * blockDim.x + threadIdx.x;
    if (idx >= NROW * HID) return;
    unsigned h1 = pcg_hash((unsigned)idx * 2u + 1u + (unsigned)samp * 0x85ebca6bu);
    unsigned h2 = pcg_hash((unsigned)idx * 2u + 2u + (unsigned)samp * 0xc2b2ae35u);
    float u1 = ((float)(h1 >> 8) + 1.0f) * (1.0f / 16777216.0f);   // (0,1]
    float u2 = (float)(h2 >> 8) * (1.0f / 16777216.0f);
    float nrm = sqrtf(-2.0f * logf(u1)) * cosf(TWO_PI_F * u2);
    z[idx] = dlt[idx] * nrm + miu[idx];
}

__global__ void mulew_kernel(const float* __restrict__ a, const float* __restrict__ b,
                             float* __restrict__ o, int n)
{
    int idx = blockIdx.x * blockDim.x + threadIdx.x;
    if (idx < n) o[idx] = a[idx] * b[idx];
}

// Y_i = Hxz @ fc5^T + b ; also stores into Y_path output slot
__global__ void fc5_kernel(const float* __restrict__ hxz,
                           const float* __restrict__ w, const float* __restrict__ b,
                           float* __restrict__ yi, float* __restrict__ ypath)
{
    int idx = blockIdx.x * blockDim.x + threadIdx.x;
    if (idx >= TYY * NROW) return;
    const float* h = hxz + (size_t)idx * HID;
    float y0 = b[0], y1 = b[1];
#pragma unroll
    for (int c = 0; c < HID; ++c) {
        y0 += h[c] * w[c];
        y1 += h[c] * w[HID + c];
    }
    yi[(size_t)idx * 2 + 0] = y0;  ypath[(size_t)idx * 2 + 0] = y0;
    yi[(size_t)idx * 2 + 1] = y1;  ypath[(size_t)idx * 2 + 1] = y1;
}

// vel = (Y_i - prev)*HZ ; Y_fv = relu(vel @ fcv^T + b)
__global__ void velfv_kernel(const float* __restrict__ yi, const float* __restrict__ cur,
                             const float* __restrict__ w, const float* __restrict__ b,
                             float* __restrict__ yfv)
{
    int idx = blockIdx.x * blockDim.x + threadIdx.x;
    if (idx >= TYY * NROW) return;
    int t = idx / NROW, n = idx % NROW;
    float px = t ? yi[(((size_t)(t - 1)) * NROW + n) * 2 + 0] : cur[n * 2 + 0];
    float py = t ? yi[(((size_t)(t - 1)) * NROW + n) * 2 + 1] : cur[n * 2 + 1];
    float vx = (yi[(size_t)idx * 2 + 0] - px) * HZ_F;
    float vy = (yi[(size_t)idx * 2 + 1] - py) * HZ_F;
#pragma unroll
    for (int c = 0; c < FVC; ++c)
        yfv[(size_t)idx * FVC + c] = fmaxf(vx * w[c * 2] + vy * w[c * 2 + 1] + b[c], 0.0f);
}

// Social pooling: one block (64 threads) per agent row n=(b,j); builds sps row.
__global__ __launch_bounds__(64)
void pool_kernel(const float* __restrict__ yi_t, const float* __restrict__ hprev,
                 float* __restrict__ sps)
{
    __shared__ float s_loc[NA * 2];
    __shared__ int   s_bin[NA];
    __shared__ float s_cnt[SPK];
    __shared__ float s_sp[SPK * HID];

    int n = blockIdx.x, b = n / NA, j = n % NA;
    int tid = threadIdx.x;

    if (tid < NA) {
        s_loc[tid * 2 + 0] = yi_t[((size_t)b * NA + tid) * 2 + 0];
        s_loc[tid * 2 + 1] = yi_t[((size_t)b * NA + tid) * 2 + 1];
    }
    __syncthreads();
    if (tid < NA) {
        float dx = s_loc[tid * 2 + 0] - s_loc[j * 2 + 0];
        float dy = s_loc[tid * 2 + 1] - s_loc[j * 2 + 1];
        float dist = sqrtf(dx * dx + dy * dy);
        int bin = -1;
        if (tid != j && dist >= 0.5f && dist <= 4.0f) {
            float th = atan2f(dy, dx);
            if (th < 0.0f) th += TWO_PI_F;
            int u = (int)floorf((dist - 0.5f) / R_STEP_F);
            u = u < 0 ? 0 : (u > 5 ? 5 : u);
            int v = (int)floorf(th / T_STEP_F);
            v = v < 0 ? 0 : (v > 5 ? 5 : v);
            bin = u * 6 + v;
        }
        s_bin[tid] = bin;
    }
    for (int idx = tid; idx < SPK * HID; idx += 64) s_sp[idx] = 0.0f;
    __syncthreads();

    if (tid < SPK) {
        int c = 0;
        for (int t2 = 0; t2 < NA; ++t2) c += (s_bin[t2] == tid);
        s_cnt[tid] = (float)(c > 1 ? c : 1);
    }
    if (tid < HID && hprev) {
        const float* hb = hprev + (size_t)b * NA * HID;
        for (int t2 = 0; t2 < NA; ++t2) {
            int bn = s_bin[t2];
            if (bn >= 0) s_sp[bn * HID + tid] += hb[(size_t)t2 * HID + tid];
        }
    }
    __syncthreads();

    float* out = sps + (size_t)n * SPD;
    for (int idx = tid; idx < SPD; idx += 64)
        out[idx] = s_sp[idx] / s_cnt[idx / HID];
}

// x_i[:, 0:32] = fm_t gather, x_i[:, 32:48] = Y_fv   (cols 48..95 filled by GEMM)
__global__ void gather_kernel(const float* __restrict__ yi_t, const float* __restrict__ yfv_t,
                              const float* __restrict__ fmt, float* __restrict__ xi)
{
    int idx = blockIdx.x * blockDim.x + threadIdx.x;
    if (idx >= NROW * HID) return;
    int n = idx / HID, c = idx % HID;
    int b = n / NA;
    float px = yi_t[(size_t)n * 2 + 0];
    float py = yi_t[(size_t)n * 2 + 1];
    int ui = 40 - (int)truncf(py);  ui = ui < 0 ? 0 : (ui > 79 ? 79 : ui);
    int vi = (int)truncf(px);       vi = vi < 0 ? 0 : (vi > 79 ? 79 : vi);
    float v;
    if (c < 32) v = fmt[(((size_t)b * 80 + ui) * 80 + vi) * 32 + c];
    else        v = yfv_t[(size_t)n * FVC + (c - 32)];
    xi[(size_t)n * 96 + c] = v;
}

// sc[n] = sum_t ( scf[t][n] . w + b )
__global__ void sc_kernel(const float* __restrict__ scf, const float* __restrict__ w,
                          const float* __restrict__ b, float* __restrict__ out)
{
    int n = blockIdx.x * blockDim.x + threadIdx.x;
    if (n >= NROW) return;
    float s = 0.0f;
    for (int t = 0; t < TYY; ++t) {
        const float* h = scf + ((size_t)t * NROW + n) * HID;
        float d = b[0];
#pragma unroll
        for (int c = 0; c < HID; ++c) d += h[c] * w[c];
        s += d;
    }
    out[n] = s;
}

// dY[t][n][d] = fcd_out[n][d*TY + t]
__global__ void dy_kernel(const float* __restrict__ fcd, float* __restrict__ out)
{
    int idx = blockIdx.x * blockDim.x + threadIdx.x;
    if (idx >= TYY * NROW * 2) return;
    int d = idx & 1, n = (idx >> 1) % NROW, t = idx / (NROW * 2);
    out[idx] = fcd[(size_t)n * 80 + d * TYY + t];
}

// ---------------------------------------------------------------------------
extern "C" void kernel_launch(void* const* d_in, const int* in_sizes, int n_in,
                              void* d_out, int out_size, void* d_ws, size_t ws_size,
                              hipStream_t stream)
{
    const float* tx  = (const float*)d_in[0];
    const float* ty  = (const float*)d_in[1];
    const float* img = (const float*)d_in[2];
    // params in setup_inputs() dict insertion order
    const float* c1w = (const float*)d_in[3];   const float* c1b = (const float*)d_in[4];
    const float* c2w = (const float*)d_in[5];   const float* c2b = (const float*)d_in[6];
    const float* e1w = (const float*)d_in[7];   const float* e1b = (const float*)d_in[8];
    const float* e2w = (const float*)d_in[9];   const float* e2b = (const float*)d_in[10];
    const float* g1wi = (const float*)d_in[11]; const float* g1wh = (const float*)d_in[12];
    const float* g1bi = (const float*)d_in[13]; const float* g1bh = (const float*)d_in[14];
    const float* g2wi = (const float*)d_in[15]; const float* g2wh = (const float*)d_in[16];
    const float* g2bi = (const float*)d_in[17]; const float* g2bh = (const float*)d_in[18];
    const float* dwi = (const float*)d_in[19];  const float* dwh = (const float*)d_in[20];
    const float* dbi = (const float*)d_in[21];  const float* dbh = (const float*)d_in[22];
    const float* f1w = (const float*)d_in[23];  const float* f1b = (const float*)d_in[24];
    const float* f2w = (const float*)d_in[25];  const float* f2b = (const float*)d_in[26];
    const float* f3w = (const float*)d_in[27];  const float* f3b = (const float*)d_in[28];
    const float* f4w = (const float*)d_in[29];  const float* f4b = (const float*)d_in[30];
    const float* f5w = (const float*)d_in[31];  const float* f5b = (const float*)d_in[32];
    const float* fvw = (const float*)d_in[33];  const float* fvb = (const float*)d_in[34];
    const float* fsw = (const float*)d_in[35];  const float* fsb = (const float*)d_in[36];
    const float* cwi = (const float*)d_in[37];  const float* cwh = (const float*)d_in[38];
    const float* cbi = (const float*)d_in[39];  const float* cbh = (const float*)d_in[40];
    const float* fdw = (const float*)d_in[41];  const float* fdb = (const float*)d_in[42];
    const float* fscw = (const float*)d_in[43]; const float* fscb = (const float*)d_in[44];

    float* out = (float*)d_out;
    float* o_yp  = out;                     // (4,40,2048,2)
    float* o_dy  = out + 655360;            // (4,40,2048,2)
    float* o_sc  = out + 1310720;           // (4,2048,1)
    float* o_miu = out + 1318912;           // (2048,48)
    float* o_dlt = out + 1417216;           // (2048,48)

    float* ws = (float*)d_ws;
    float* fmt  = ws;                       // 6,553,600
    float* ex   = fmt  + 6553600;           //   655,360
    float* ey   = ex   + 655360;            // 1,310,720
    float* hx   = ey   + 1310720;           //    98,304
    float* hy   = hx   + 98304;
    float* hcat = hy   + 98304;             //   196,608
    float* hc   = hcat + 196608;            //    98,304
    float* zb   = hc   + 98304;
    float* beta = zb   + 98304;
    float* cur  = beta + 98304;             //     4,096
    float* dyn  = cur  + 4096;              // dynamic region
    float* fm1  = dyn;                      // 13,107,200 (conv phase only)
    float* hxz  = dyn;                      //  3,932,160 (per-sample phase)
    float* yi   = hxz + 3932160;            //    163,840
    float* yfv  = yi  + 163840;             //  1,310,720
    float* scf  = yfv + 1310720;            //  3,932,160
    float* sps  = scf + 3932160;            //  3,538,944
    float* xi   = sps + 3538944;            //    196,608
    float* fcd  = xi  + 196608;             //    163,840

    dim3 b256(256);

    // ---- image feature stack ----
    conv1_kernel<<<3200, b256, 0, stream>>>(img, c1w, c1b, fm1);
    conv2_kernel<<<800,  b256, 0, stream>>>(fm1, c2w, c2b, fmt);
    cur_kernel<<<16, b256, 0, stream>>>(tx, cur);

    // ---- trajectory encoders ----
    enc1d_kernel<<<2560, b256, 0, stream>>>(tx, TXX, e1w, e1b, ex);
    enc1d_kernel<<<5120, b256, 0, stream>>>(ty, TYY, e2w, e2b, ey);
    for (int t = 0; t < TXX; ++t)
        gru_step_wmma<<<NROW / 16, 288, 0, stream>>>(
            ex + (size_t)t * NROW * CE, CE, t ? hx : nullptr,
            g1wi, g1wh, g1bi, g1bh, hx);
    for (int t = 0; t < TYY; ++t)
        gru_step_wmma<<<NROW / 16, 288, 0, stream>>>(
            ey + (size_t)t * NROW * CE, CE, t ? hy : nullptr,
            g2wi, g2wh, g2bi, g2bh, hy);

    // ---- latent heads ----
    concat_kernel<<<768, b256, 0, stream>>>(hx, hy, hcat);
    gemm_wmma_f32<<<dim3(NROW / 16, 3), 32, 0, stream>>>(hcat, 96, f1w, 96, f1b, hc, HID, 96, 1);
    gemm_wmma_f32<<<dim3(NROW / 16, 3), 32, 0, stream>>>(hc, HID, f2w, HID, f2b, o_miu, HID, HID, 0);
    gemm_wmma_f32<<<dim3(NROW / 16, 3), 32, 0, stream>>>(hc, HID, f3w, HID, f3b, o_dlt, HID, HID, 2);

    // ---- K samples ----
    for (int i = 0; i < KSAMP; ++i) {
        znoise_kernel<<<384, b256, 0, stream>>>(o_miu, o_dlt, zb, i);
        gemm_wmma_f32<<<dim3(NROW / 16, 3), 32, 0, stream>>>(zb, HID, f4w, HID, f4b, beta, HID, HID, 0);
        mulew_kernel<<<384, b256, 0, stream>>>(hx, beta, beta, NROW * HID); // x0 = new_Hx*beta

        // decoder GRU (x only at t=0)
        for (int t = 0; t < TYY; ++t)
            gru_step_wmma<<<NROW / 16, 288, 0, stream>>>(
                t == 0 ? beta : nullptr, HID,
                t ? hxz + (size_t)(t - 1) * NROW * HID : nullptr,
                dwi, dwh, dbi, dbh, hxz + (size_t)t * NROW * HID);

        fc5_kernel<<<320, b256, 0, stream>>>(hxz, f5w, f5b, yi, o_yp + (size_t)i * 163840);
        velfv_kernel<<<320, b256, 0, stream>>>(yi, cur, fvw, fvb, yfv);

        // social refinement scan (ITERS = 1)
        for (int t = 0; t < TYY; ++t) {
            const float* hprev = t ? scf + (size_t)(t - 1) * NROW * HID : nullptr;
            pool_kernel<<<NROW, 64, 0, stream>>>(yi + (size_t)t * NROW * 2, hprev, sps);
            gemm_fcs_wmma<<<dim3(NROW / 16, 3), 32, 0, stream>>>(
                sps, SPD, fsw, SPD, fsb, xi + 48, 96, SPD);             // rhalf -> xi[:,48:96]
            gather_kernel<<<384, b256, 0, stream>>>(
                yi + (size_t)t * NROW * 2, yfv + (size_t)t * NROW * FVC, fmt, xi);
            gru_step_wmma<<<NROW / 16, 288, 0, stream>>>(
                xi, 96, hprev, cwi, cwh, cbi, cbh, scf + (size_t)t * NROW * HID);
        }

        gemm_wmma_f32<<<dim3(NROW / 16, 5), 32, 0, stream>>>(
            scf + (size_t)(TYY - 1) * NROW * HID, HID, fdw, HID, fdb, fcd, 80, HID, 0);
        dy_kernel<<<640, b256, 0, stream>>>(fcd, o_dy + (size_t)i * 163840);
        sc_kernel<<<8, b256, 0, stream>>>(scf, fscw, fscb, o_sc + (size_t)i * NROW);
    }
}